// ContinuousFilterConv_60361470378647
// MI455X (gfx1250) — compile-verified
//
#include <hip/hip_runtime.h>
#include <math.h>

#define HIDDEN 64
#define CUTOFF 5.0f

typedef __attribute__((ext_vector_type(2))) float v2f;
typedef __attribute__((ext_vector_type(8))) float v8f;

// -------------------- kernel: zero the aggregation buffer --------------------
__global__ void cfconv_zero(float4* __restrict__ p, int n4) {
    int i = blockIdx.x * blockDim.x + threadIdx.x;
    if (i < n4) p[i] = make_float4(0.f, 0.f, 0.f, 0.f);
}

// -------------------- kernel: rowsum(fw2) and sum(fb2) --------------------
// w2s[0..63] = sum_f fw2[k,f];  w2s[64] = sum_f fb2[f]
__global__ void cfconv_prep(const float* __restrict__ fw2,
                            const float* __restrict__ fb2,
                            float* __restrict__ w2s) {
    int k = threadIdx.x;  // 64 threads
    float s = 0.f;
    for (int f = 0; f < HIDDEN; ++f) s += fw2[k * HIDDEN + f];
    w2s[k] = s;
    if (k == 0) {
        float b = 0.f;
        for (int f = 0; f < HIDDEN; ++f) b += fb2[f];
        w2s[HIDDEN] = b;
    }
}

// -------------------- kernel: per-edge filter + gather/scatter --------------------
// One wave32 per edge. Lane l covers hidden dims {2l, 2l+1} (coalesced 256B/edge).
// Filter dot computed cooperatively: lane l handles filter indices l and l+32.
__global__ void cfconv_edges(const float* __restrict__ x,
                             const int* __restrict__ eidx,   // [2,E] flat
                             const float* __restrict__ dist,
                             const float* __restrict__ fw1,
                             const float* __restrict__ fb1,
                             const float* __restrict__ w2s,
                             float* __restrict__ agg, int E) {
    const int lane = threadIdx.x & 31;
    const int wid  = blockIdx.x * (blockDim.x >> 5) + (threadIdx.x >> 5);
    const int nw   = gridDim.x * (blockDim.x >> 5);

    // lane-resident filter-net params (read once, L2/SGPR friendly)
    const float w1a = fw1[lane],      w1b = fw1[lane + 32];
    const float b1a = fb1[lane],      b1b = fb1[lane + 32];
    const float s2a = w2s[lane],      s2b = w2s[lane + 32];
    const float bsum = w2s[HIDDEN];
    const float inv  = 2.0f / CUTOFF;
    const float pic  = 3.14159265358979f / CUTOFF;

    for (int e = wid; e < E; e += nw) {
        const int row = eidx[e];           // wave-uniform -> scalar load
        const int col = eidx[E + e];
        const float d = dist[e];
        const float scaled = fmaf(d, inv, -1.0f);
        float t = tanhf(fmaf(scaled, w1a, b1a)) * s2a
                + tanhf(fmaf(scaled, w1b, b1b)) * s2b;
        #pragma unroll
        for (int o = 16; o > 0; o >>= 1) t += __shfl_xor(t, o, 32);
        float cut = (d <= CUTOFF) ? 0.5f * (cosf(d * pic) + 1.0f) : 0.0f;
        const float s = (t + bsum) * cut;

        const float2 v = *(const float2*)(x + (size_t)col * HIDDEN + 2 * lane);
        float* ap = agg + (size_t)row * HIDDEN + 2 * lane;
        atomicAdd(ap,     v.x * s);
        atomicAdd(ap + 1, v.y * s);
    }
}

// -------------------- WMMA helper --------------------
__device__ __forceinline__ v8f wmma4(v2f a, v2f b, v8f c) {
    // D(16x16,f32) = A(16x4,f32) x B(4x16,f32) + C
    return __builtin_amdgcn_wmma_f32_16x16x4_f32(
        /*neg_a=*/false, a, /*neg_b=*/false, b,
        /*c_mod=*/(short)0, c, /*reuse_a=*/false, /*reuse_b=*/false);
}

// -------------------- kernel: node MLP, out = softplus(agg@iw1+ib1)@iw2+ib2 ----
// 8 waves/block; each wave owns a 16-row node tile. K=64 -> 16 chained WMMAs per
// 16-col output tile, x4 col tiles, x2 stages = 128 v_wmma per wave.
// Stage-1 result (D layout) round-trips through a 4KB per-wave LDS tile so it
// can be re-read in A-fragment layout for stage 2 (same-wave DS is in-order).
__global__ void cfconv_mlp(const float* __restrict__ agg,
                           const float* __restrict__ iw1, const float* __restrict__ ib1,
                           const float* __restrict__ iw2, const float* __restrict__ ib2,
                           float* __restrict__ out, int N) {
    __shared__ float lds[8 * 16 * HIDDEN];  // 32 KB
    const int lane = threadIdx.x & 31;
    const int wave = threadIdx.x >> 5;
    const int n    = lane & 15;      // output column within 16-col tile
    const int hi   = lane >> 4;      // half-wave select (K pairing / M+8)
    float* tile = lds + wave * 16 * HIDDEN;
    const int rows0 = (blockIdx.x * 8 + wave) * 16;

    // ---- stage 1: h = agg @ iw1 + ib1 ----
    v8f acc[4];
    #pragma unroll
    for (int j = 0; j < 4; ++j) {
        const float b = ib1[j * 16 + n];
        #pragma unroll
        for (int r = 0; r < 8; ++r) acc[j][r] = b;
    }
    const int arow = rows0 + n;             // A-fragment row for this lane
    const bool rowOK = (arow < N);
    #pragma unroll
    for (int k0 = 0; k0 < HIDDEN; k0 += 4) {
        const int kk = k0 + 2 * hi;         // lanes 0-15: K=k0,k0+1; 16-31: k0+2,k0+3
        v2f a;
        if (rowOK) {
            const float* ap = agg + (size_t)arow * HIDDEN + kk;
            a.x = ap[0]; a.y = ap[1];
        } else { a.x = 0.f; a.y = 0.f; }
        #pragma unroll
        for (int j = 0; j < 4; ++j) {
            v2f b;
            b.x = iw1[(kk)     * HIDDEN + j * 16 + n];
            b.y = iw1[(kk + 1) * HIDDEN + j * 16 + n];
            acc[j] = wmma4(a, b, acc[j]);
        }
    }

    // ---- softplus, spill 16x64 tile to LDS in row-major ----
    #pragma unroll
    for (int j = 0; j < 4; ++j) {
        #pragma unroll
        for (int r = 0; r < 8; ++r) {
            const int M = r + 8 * hi;       // D layout: vgpr r -> row r (+8 for upper half)
            const float xv = acc[j][r];
            const float sp = fmaxf(xv, 0.f) + log1pf(expf(-fabsf(xv)));
            tile[M * HIDDEN + j * 16 + n] = sp;
        }
    }

    // ---- stage 2: out = h @ iw2 + ib2 ----
    v8f acc2[4];
    #pragma unroll
    for (int j = 0; j < 4; ++j) {
        const float b = ib2[j * 16 + n];
        #pragma unroll
        for (int r = 0; r < 8; ++r) acc2[j][r] = b;
    }
    #pragma unroll
    for (int k0 = 0; k0 < HIDDEN; k0 += 4) {
        const int kk = k0 + 2 * hi;
        v2f a;                               // A fragment from LDS tile (row n)
        a.x = tile[n * HIDDEN + kk];
        a.y = tile[n * HIDDEN + kk + 1];
        #pragma unroll
        for (int j = 0; j < 4; ++j) {
            v2f b;
            b.x = iw2[(kk)     * HIDDEN + j * 16 + n];
            b.y = iw2[(kk + 1) * HIDDEN + j * 16 + n];
            acc2[j] = wmma4(a, b, acc2[j]);
        }
    }

    // ---- store D tiles with row guard ----
    #pragma unroll
    for (int j = 0; j < 4; ++j) {
        #pragma unroll
        for (int r = 0; r < 8; ++r) {
            const int g = rows0 + r + 8 * hi;
            if (g < N) out[(size_t)g * HIDDEN + j * 16 + n] = acc2[j][r];
        }
    }
}

extern "C" void kernel_launch(void* const* d_in, const int* in_sizes, int n_in,
                              void* d_out, int out_size, void* d_ws, size_t ws_size,
                              hipStream_t stream) {
    const float* x    = (const float*)d_in[0];
    const int*   eidx = (const int*)  d_in[1];
    const float* dist = (const float*)d_in[2];
    /* d_in[3] = edge_attr (unused by the layer) */
    const float* fw1  = (const float*)d_in[4];
    const float* fb1  = (const float*)d_in[5];
    const float* fw2  = (const float*)d_in[6];
    const float* fb2  = (const float*)d_in[7];
    const float* iw1  = (const float*)d_in[8];
    const float* ib1  = (const float*)d_in[9];
    const float* iw2  = (const float*)d_in[10];
    const float* ib2  = (const float*)d_in[11];
    float* out = (float*)d_out;

    const int N = in_sizes[0] / HIDDEN;   // 100000
    const int E = in_sizes[2];            // 3200000

    // workspace: agg [N*64 f32] then w2s [65 f32]
    float* agg = (float*)d_ws;
    float* w2s = agg + (size_t)N * HIDDEN;

    const int n4 = N * HIDDEN / 4;
    cfconv_zero<<<(n4 + 255) / 256, 256, 0, stream>>>((float4*)agg, n4);
    cfconv_prep<<<1, 64, 0, stream>>>(fw2, fb2, w2s);
    cfconv_edges<<<1920, 256, 0, stream>>>(x, eidx, dist, fw1, fb1, w2s, agg, E);
    const int nb = (N + 127) / 128;       // 128 rows per block (8 waves x 16)
    cfconv_mlp<<<nb, 256, 0, stream>>>(agg, iw1, ib1, iw2, ib2, out, N);
}